// PhysicalRIRModel_69930657513512
// MI455X (gfx1250) — compile-verified
//
#include <hip/hip_runtime.h>
#include <math.h>

// ---------------------------------------------------------------------------
// PhysicalRIRModel.lvecmul_PEG on gfx1250 (MI455X), fp32 WMMA formulation.
//   Lh = 8192, Lg = 128, Lp = 16, Nfft = 16384
// Step A: w = conv(x,g);  xg[n] = w[n] + xg[n-2]   (parity prefix scan)
// Step B: Yf[k] = sum_n xg[n] v[k]^n,  v = FFT(p e^a) .* omega
//         block split n = 64q+r  ->  Inner = X(128x64) @ V(64x16384)  [WMMA]
//         Yf = Horner_q(Inner; v^64)                                  [LDS]
// Step C: y[n] = (1/N) Re sum_k Yf[k] e^{2pi i kn/N}
//         block split k = 128q+r -> Inner2 = Yf(128x128) @ P(128x8192)[WMMA]
//         y = (1/N) Re Horner_q(Inner2; e^{2pi i n/128})              [LDS]
// ---------------------------------------------------------------------------

#define LH    8192
#define LG    128
#define LP    16
#define NFFT  16384
#define TWO_PI 6.28318530717958647692f

typedef __attribute__((ext_vector_type(2))) float v2f;
typedef __attribute__((ext_vector_type(8))) float v8f;

__device__ __forceinline__ float2 cmul(float2 a, float2 b) {
    return make_float2(a.x * b.x - a.y * b.y, a.x * b.y + a.y * b.x);
}

__device__ __forceinline__ v8f wmma4(v2f a, v2f b, v8f c) {
    // V_WMMA_F32_16X16X4_F32 : D(16x16,f32) = A(16x4,f32) x B(4x16,f32) + C
    return __builtin_amdgcn_wmma_f32_16x16x4_f32(
        /*neg_a=*/false, a, /*neg_b=*/false, b,
        /*c_mod=*/(short)0, c, /*reuse_a=*/false, /*reuse_b=*/false);
}

// ---------------------------------------------------------------------------
// Kernel 1a: w = conv(x, g)[:Lh], then xg = stride-2 parity prefix scan of w.
// Single workgroup of 1024 threads, 8 elements/thread, LDS Hillis-Steele.
// ---------------------------------------------------------------------------
__global__ __launch_bounds__(1024)
void k_conv_scan(const float* __restrict__ x, const float* __restrict__ g,
                 float* __restrict__ xg) {
    __shared__ float buf[LH];
    const int t = threadIdx.x;

    #pragma unroll
    for (int i = 0; i < 8; ++i) {
        const int n = i * 1024 + t;
        float acc = 0.f;
        const int mmax = n < (LG - 1) ? n : (LG - 1);
        for (int m = 0; m <= mmax; ++m)
            acc += g[m] * x[n - m];
        buf[n] = acc;
    }
    __syncthreads();

    // inclusive scan over each parity class: offsets 2,4,...,4096
    for (int d = 2; d < LH; d <<= 1) {
        float tmp[8];
        #pragma unroll
        for (int i = 0; i < 8; ++i) {
            const int n = i * 1024 + t;
            tmp[i] = buf[n] + ((n >= d) ? buf[n - d] : 0.f);
        }
        __syncthreads();
        #pragma unroll
        for (int i = 0; i < 8; ++i)
            buf[i * 1024 + t] = tmp[i];
        __syncthreads();
    }

    #pragma unroll
    for (int i = 0; i < 8; ++i) {
        const int n = i * 1024 + t;
        xg[n] = buf[n];
    }
}

// ---------------------------------------------------------------------------
// Kernel 1b: v[k] = e^a * (sum_m p[m] e^{-2pi i km/N}) * e^{-2pi i k/N}
// ---------------------------------------------------------------------------
__global__ __launch_bounds__(256)
void k_twiddle(const float* __restrict__ p, const float* __restrict__ a,
               float2* __restrict__ v) {
    const int k = blockIdx.x * 256 + threadIdx.x;
    const float ea = __expf(a[0]);
    float re = 0.f, im = 0.f;
    #pragma unroll
    for (int m = 0; m < LP; ++m) {
        const int km = (k * m) & (NFFT - 1);        // exact arg reduction
        float s, c;
        __sincosf(-TWO_PI * (float)km / (float)NFFT, &s, &c);
        re += p[m] * c;
        im += p[m] * s;
    }
    float s, c;
    __sincosf(-TWO_PI * (float)k / (float)NFFT, &s, &c);
    v[k] = cmul(make_float2(ea * re, ea * im), make_float2(c, s));
}

// ---------------------------------------------------------------------------
// Kernel 2: Yf[k] = Horner_q( (X @ V)[q][k] ; v[k]^64 )
// One wave per 16-wide k tile; 8 M-tiles (q=0..127) held in registers;
// V[r][k] = v[k]^r generated on the fly by per-lane complex power stepping.
// Block = 2 waves.
// ---------------------------------------------------------------------------
__global__ __launch_bounds__(64)
void k_polyeval_wmma(const float* __restrict__ xg, const float2* __restrict__ v,
                     float2* __restrict__ Yf) {
    __shared__ float lre[2][128][16];
    __shared__ float lim[2][128][16];

    const int wave = threadIdx.x >> 5;
    const int lane = threadIdx.x & 31;
    const int col  = lane & 15;
    const int half = lane >> 4;
    const int k    = (blockIdx.x * 2 + wave) * 16 + col;

    const float2 v1 = v[k];
    const float2 v2 = cmul(v1, v1);
    const float2 v4 = cmul(v2, v2);
    float2 pc = half ? v2 : make_float2(1.f, 0.f);   // v^(4t + 2*half)

    v8f accRe[8], accIm[8];
    #pragma unroll
    for (int m = 0; m < 8; ++m) { accRe[m] = {}; accIm[m] = {}; }

    for (int t = 0; t < 16; ++t) {                   // K = 64 = 16 * 4
        const float2 b0 = pc;
        const float2 b1 = cmul(pc, v1);
        const v2f bRe = {b0.x, b1.x};
        const v2f bIm = {b0.y, b1.y};
        const int rb = 4 * t + 2 * half;
        #pragma unroll
        for (int m = 0; m < 8; ++m) {
            const int q = m * 16 + col;              // A rows, M-tile m
            const v2f afr = {xg[64 * q + rb], xg[64 * q + rb + 1]};
            accRe[m] = wmma4(afr, bRe, accRe[m]);
            accIm[m] = wmma4(afr, bIm, accIm[m]);
        }
        pc = cmul(pc, v4);
    }

    #pragma unroll
    for (int m = 0; m < 8; ++m) {
        #pragma unroll
        for (int i = 0; i < 8; ++i) {
            const int q = m * 16 + i + 8 * half;     // C-tile row mapping
            lre[wave][q][col] = accRe[m][i];
            lim[wave][q][col] = accIm[m][i];
        }
    }
    __syncthreads();

    // u = v^64 ; Horner over q = 127..0
    float2 u = v4;
    u = cmul(u, u); u = cmul(u, u); u = cmul(u, u); u = cmul(u, u);
    float2 acc = make_float2(0.f, 0.f);
    for (int q = 127; q >= 0; --q) {
        acc = cmul(acc, u);
        acc.x += lre[wave][q][col];
        acc.y += lim[wave][q][col];
    }
    if (half == 0) Yf[k] = acc;
}

// ---------------------------------------------------------------------------
// Kernel 3: y[n] = (1/N) Re Horner_q( (Yf @ P)[q][n] ; e^{2pi i n/128} )
// Complex x complex matmul: 4 WMMA chains per M-tile.
// ---------------------------------------------------------------------------
__global__ __launch_bounds__(64)
void k_idft_wmma(const float2* __restrict__ Yf, float* __restrict__ y) {
    __shared__ float lre[2][128][16];
    __shared__ float lim[2][128][16];

    const int wave = threadIdx.x >> 5;
    const int lane = threadIdx.x & 31;
    const int col  = lane & 15;
    const int half = lane >> 4;
    const int n    = (blockIdx.x * 2 + wave) * 16 + col;

    float s, c;
    __sincosf(TWO_PI * (float)n / (float)NFFT, &s, &c);
    const float2 W  = make_float2(c, s);             // e^{+2pi i n/N}
    const float2 W2 = cmul(W, W);
    const float2 W4 = cmul(W2, W2);
    float2 pc = half ? W2 : make_float2(1.f, 0.f);

    v8f Cr[8], Ci[8];
    #pragma unroll
    for (int m = 0; m < 8; ++m) { Cr[m] = {}; Ci[m] = {}; }

    for (int t = 0; t < 32; ++t) {                   // K = 128 = 32 * 4
        const float2 b0 = pc;
        const float2 b1 = cmul(pc, W);
        const v2f bRe  = {b0.x, b1.x};
        const v2f bIm  = {b0.y, b1.y};
        const v2f bImN = {-b0.y, -b1.y};
        const int rb = 4 * t + 2 * half;
        #pragma unroll
        for (int m = 0; m < 8; ++m) {
            const int q = m * 16 + col;
            const float2 a0 = Yf[128 * q + rb];
            const float2 a1 = Yf[128 * q + rb + 1];
            const v2f aRe = {a0.x, a1.x};
            const v2f aIm = {a0.y, a1.y};
            Cr[m] = wmma4(aRe, bRe,  Cr[m]);         // +ReA*ReB
            Cr[m] = wmma4(aIm, bImN, Cr[m]);         // -ImA*ImB
            Ci[m] = wmma4(aRe, bIm,  Ci[m]);         // +ReA*ImB
            Ci[m] = wmma4(aIm, bRe,  Ci[m]);         // +ImA*ReB
        }
        pc = cmul(pc, W4);
    }

    #pragma unroll
    for (int m = 0; m < 8; ++m) {
        #pragma unroll
        for (int i = 0; i < 8; ++i) {
            const int q = m * 16 + i + 8 * half;
            lre[wave][q][col] = Cr[m][i];
            lim[wave][q][col] = Ci[m][i];
        }
    }
    __syncthreads();

    // U = W^128 = e^{2pi i n/128}
    __sincosf(TWO_PI * (float)(n & 127) / 128.f, &s, &c);
    const float2 U = make_float2(c, s);
    float2 acc = make_float2(0.f, 0.f);
    for (int q = 127; q >= 0; --q) {
        acc = cmul(acc, U);
        acc.x += lre[wave][q][col];
        acc.y += lim[wave][q][col];
    }
    if (half == 0) y[n] = acc.x * (1.f / (float)NFFT);
}

// ---------------------------------------------------------------------------
extern "C" void kernel_launch(void* const* d_in, const int* in_sizes, int n_in,
                              void* d_out, int out_size, void* d_ws, size_t ws_size,
                              hipStream_t stream) {
    const float* x = (const float*)d_in[0];   // (8192,)
    const float* g = (const float*)d_in[1];   // (128,)
    const float* a = (const float*)d_in[2];   // scalar
    const float* p = (const float*)d_in[3];   // (16,)
    float* y = (float*)d_out;                 // (8192,)

    // workspace layout: xg (8192 f) | v (16384 c64) | Yf (16384 c64) ~ 288 KB
    float*  xg = (float*)d_ws;
    float2* v  = (float2*)(xg + LH);
    float2* Yf = v + NFFT;

    k_conv_scan<<<1, 1024, 0, stream>>>(x, g, xg);
    k_twiddle<<<NFFT / 256, 256, 0, stream>>>(p, a, v);
    k_polyeval_wmma<<<(NFFT / 16) / 2, 64, 0, stream>>>(xg, v, Yf);
    k_idft_wmma<<<(LH / 16) / 2, 64, 0, stream>>>(Yf, y);
}